// FPS_kNN_5188320494094
// MI455X (gfx1250) — compile-verified
//
#include <hip/hip_runtime.h>

#define B_ 8
#define N_ 16384
#define C_ 64
#define G_ 512
#define K_ 32

typedef __attribute__((ext_vector_type(2))) float v2f;
typedef __attribute__((ext_vector_type(8))) float v8f;

// ------------------------------------------------------------------
// Kernel 1: Furthest Point Sampling. One block per batch, 1024 threads,
// 16 points per thread held in VGPRs. Two-level argmax reduction
// (wave32 shfl_xor, then cross-wave through LDS). Matches the reference:
// emitted index at step t is `last` BEFORE the update (first index is 0).
// ------------------------------------------------------------------
__global__ __launch_bounds__(1024) void fps_kernel(const float* __restrict__ xyz,
                                                   int* __restrict__ fps_idx) {
  const int b    = blockIdx.x;
  const int tid  = threadIdx.x;
  const int lane = tid & 31;
  const int wid  = tid >> 5;
  const float* bxyz = xyz + (size_t)b * N_ * 3;

  __shared__ int   s_last;
  __shared__ float s_rv[32];
  __shared__ int   s_ri[32];

  float px[16], py[16], pz[16], md[16];
#pragma unroll
  for (int i = 0; i < 16; ++i) {
    int n = i * 1024 + tid;
    px[i] = bxyz[n * 3 + 0];
    py[i] = bxyz[n * 3 + 1];
    pz[i] = bxyz[n * 3 + 2];
    md[i] = 1e10f;
  }
  if (tid == 0) s_last = 0;
  __syncthreads();

  for (int t = 0; t < G_; ++t) {
    int last = s_last;
    if (tid == 0) fps_idx[b * G_ + t] = last;
    // broadcast coords of newest selected point (uniform address, L0/L2 hit)
    float sx = bxyz[last * 3 + 0];
    float sy = bxyz[last * 3 + 1];
    float sz = bxyz[last * 3 + 2];

    float best = -1.0f;
    int   bidx = 0;
#pragma unroll
    for (int i = 0; i < 16; ++i) {
      float dx = px[i] - sx, dy = py[i] - sy, dz = pz[i] - sz;
      float d  = dx * dx + dy * dy + dz * dz;
      float m  = fminf(md[i], d);
      md[i] = m;
      int n = i * 1024 + tid;
      if (m > best) { best = m; bidx = n; }   // first-occurrence (smallest n) on ties
    }
    // wave32 argmax reduce, ties -> smaller index (jnp.argmax semantics)
#pragma unroll
    for (int off = 16; off >= 1; off >>= 1) {
      float ov = __shfl_xor(best, off, 32);
      int   oi = __shfl_xor(bidx, off, 32);
      if (ov > best || (ov == best && oi < bidx)) { best = ov; bidx = oi; }
    }
    if (lane == 0) { s_rv[wid] = best; s_ri[wid] = bidx; }
    __syncthreads();
    if (wid == 0) {
      best = s_rv[lane];
      bidx = s_ri[lane];
#pragma unroll
      for (int off = 16; off >= 1; off >>= 1) {
        float ov = __shfl_xor(best, off, 32);
        int   oi = __shfl_xor(bidx, off, 32);
        if (ov > best || (ov == best && oi < bidx)) { best = ov; bidx = oi; }
      }
      if (lane == 0) s_last = bidx;
    }
    __syncthreads();
  }
}

// ------------------------------------------------------------------
// Kernel 2: gather lc_xyz [B,G,3] and lc_x [B,G,C] from fps indices.
// One 64-thread block per (b,g): thread = channel (coalesced 256B rows).
// ------------------------------------------------------------------
__global__ __launch_bounds__(64) void gather_kernel(const float* __restrict__ xyz,
                                                    const float* __restrict__ x,
                                                    const int* __restrict__ fps_idx,
                                                    float* __restrict__ lc_xyz,
                                                    float* __restrict__ lc_x) {
  int bg  = blockIdx.x;          // b*G + g
  int b   = bg / G_;
  int tid = threadIdx.x;
  int idx = fps_idx[bg];
  size_t src = (size_t)b * N_ + idx;
  lc_x[(size_t)bg * C_ + tid] = x[src * C_ + tid];
  if (tid < 3) lc_xyz[(size_t)bg * 3 + tid] = xyz[src * 3 + tid];
}

// ------------------------------------------------------------------
// Kernel 3: kNN via WMMA f32 16x16x4 + streaming per-row top-32.
// Block = (b, tile of 16 centers), 256 threads = 8 waves.
// D = A' x B + C with A' = -2*centers (16x4, K=3 padded), B = points (4x16),
// C preloaded with |c|^2 + |p|^2  ==>  D = exact squared distances.
// 256 columns per sweep iteration (two WMMAs per wave, shared A frag).
//
// Fragment layouts per CDNA5 ISA 7.12.2 (32-bit, wave32):
//   A 16x4 : lane L holds M=L%16; VGPR0 = K=(L<16?0:2), VGPR1 = K=(L<16?1:3)
//   B 4x16 : lane L holds N=L%16; VGPR0 = K=(L<16?0:2), VGPR1 = K=(L<16?1:3)
//   C/D    : lane L holds N=L%16; VGPR r -> M = r + (L<16?0:8)
//
// Selection uses PLAIN (non-volatile) LDS accesses + asm memory clobbers:
// volatile __shared__ blocks InferAddressSpaces and degrades to SYS-scope
// FLAT loads; plain accesses stay addrspace(3) -> ds_load/ds_store, and
// same-wave LDS ops are hardware in-order, so a clobber-forced re-read
// observes the leader's store.
// ------------------------------------------------------------------
__global__ __launch_bounds__(256) void knn_kernel(const float* __restrict__ xyz,
                                                  const float* __restrict__ x,
                                                  const float* __restrict__ lc_xyz,
                                                  float* __restrict__ knn_xyz,
                                                  float* __restrict__ knn_x) {
  const int b    = blockIdx.y;
  const int g0   = blockIdx.x * 16;
  const int tid  = threadIdx.x;
  const int lane = tid & 31;
  const int w    = tid >> 5;      // wave id 0..7
  const int lm   = lane & 15;
  const bool hi  = (lane >= 16);

  __shared__ float s_cx[16], s_cy[16], s_cz[16], s_cn[16];
  __shared__ float s_d[16][264];                 // 16 rows x 256 cols (+8 pad)
  __shared__ float s_topd[16][K_];
  __shared__ int   s_topi[16][K_];
  __shared__ float s_rowmax[16];
  __shared__ int   s_rowpos[16];

  const float* bxyz = xyz + (size_t)b * N_ * 3;

  if (tid < 16) {
    int g = g0 + tid;
    float cx = lc_xyz[((size_t)b * G_ + g) * 3 + 0];
    float cy = lc_xyz[((size_t)b * G_ + g) * 3 + 1];
    float cz = lc_xyz[((size_t)b * G_ + g) * 3 + 2];
    s_cx[tid] = cx; s_cy[tid] = cy; s_cz[tid] = cz;
    s_cn[tid] = cx * cx + cy * cy + cz * cz;
    s_rowmax[tid] = 1e30f;
    s_rowpos[tid] = 0;
  }
  for (int i = tid; i < 16 * K_; i += 256) {
    s_topd[i / K_][i % K_] = 1e30f;
    s_topi[i / K_][i % K_] = 0;
  }
  __syncthreads();

  // A fragment: -2 * centers, constant across the whole sweep
  v2f afrag;
  afrag.x = -2.0f * (hi ? s_cz[lm] : s_cx[lm]);
  afrag.y = hi ? 0.0f : (-2.0f * s_cy[lm]);

  for (int nb = 0; nb < N_; nb += 256) {
    // wave w computes columns nb+16w..+15 and nb+128+16w..+15
    int n0 = nb + w * 16 + lm;
    int n1 = n0 + 128;

    // prefetch the tile two sweeps ahead (global_prefetch_b8)
    {
      int npf = n0 + 512;
      if (npf < N_) __builtin_prefetch(&bxyz[(size_t)npf * 3], 0, 1);
    }

    const int mbase = hi ? 8 : 0;

    float qx0 = bxyz[n0 * 3 + 0], qy0 = bxyz[n0 * 3 + 1], qz0 = bxyz[n0 * 3 + 2];
    float pn0 = qx0 * qx0 + qy0 * qy0 + qz0 * qz0;
    v2f bfrag0;
    bfrag0.x = hi ? qz0 : qx0;
    bfrag0.y = hi ? 0.0f : qy0;
    v8f cfrag0;
#pragma unroll
    for (int r = 0; r < 8; ++r) cfrag0[r] = s_cn[mbase + r] + pn0;
    v8f dfrag0 = __builtin_amdgcn_wmma_f32_16x16x4_f32(
        false, afrag, false, bfrag0, (short)0, cfrag0, false, false);
#pragma unroll
    for (int r = 0; r < 8; ++r) s_d[mbase + r][w * 16 + lm] = dfrag0[r];

    float qx1 = bxyz[n1 * 3 + 0], qy1 = bxyz[n1 * 3 + 1], qz1 = bxyz[n1 * 3 + 2];
    float pn1 = qx1 * qx1 + qy1 * qy1 + qz1 * qz1;
    v2f bfrag1;
    bfrag1.x = hi ? qz1 : qx1;
    bfrag1.y = hi ? 0.0f : qy1;
    v8f cfrag1;
#pragma unroll
    for (int r = 0; r < 8; ++r) cfrag1[r] = s_cn[mbase + r] + pn1;
    v8f dfrag1 = __builtin_amdgcn_wmma_f32_16x16x4_f32(
        false, afrag, false, bfrag1, (short)0, cfrag1, false, false);
#pragma unroll
    for (int r = 0; r < 8; ++r) s_d[mbase + r][128 + w * 16 + lm] = dfrag1[r];

    __syncthreads();

    // Selection: wave w owns rows 2w and 2w+1 exclusively. One insertion
    // per round, lowest pending lane wins; re-reads forced by clobber.
#pragma unroll
    for (int rr = 0; rr < 2; ++rr) {
      int row = 2 * w + rr;
#pragma unroll
      for (int cc = 0; cc < 8; ++cc) {
        int   col = lane + cc * 32;
        float d   = s_d[row][col];
        int   idx = nb + col;
        while (true) {
          asm volatile("" ::: "memory");        // force fresh LDS reads
          float cm  = s_rowmax[row];            // ds_load_b32
          bool pend = d < cm;
          unsigned long long mb = __ballot(pend);
          if (!mb) break;
          int leader = __ffsll((long long)mb) - 1;
          if (lane == leader) {
            // Only writer this round: insert unconditionally, refresh max.
            int p = s_rowpos[row];
            s_topd[row][p] = d;
            s_topi[row][p] = idx;
            float mxv = -1e38f; int mxp = 0;
            for (int j = 0; j < K_; ++j) {
              float v = s_topd[row][j];
              if (v > mxv) { mxv = v; mxp = j; }
            }
            s_rowmax[row] = mxv;
            s_rowpos[row] = mxp;
            d = 1e30f;                          // consume own candidate
          }
          asm volatile("" ::: "memory");
        }
      }
    }
    __syncthreads();
  }

  // Sort each row's 32 candidates ascending (thread t handles row t)
  if (tid < 16) {
    for (int i = 0; i < K_ - 1; ++i) {
      int mi = i; float mv = s_topd[tid][i]; int mvi = s_topi[tid][i];
      for (int j = i + 1; j < K_; ++j) {
        float v = s_topd[tid][j]; int vi = s_topi[tid][j];
        if (v < mv || (v == mv && vi < mvi)) { mi = j; mv = v; mvi = vi; }
      }
      if (mi != i) {
        s_topd[tid][mi] = s_topd[tid][i]; s_topi[tid][mi] = s_topi[tid][i];
        s_topd[tid][i]  = mv;             s_topi[tid][i]  = mvi;
      }
    }
  }
  __syncthreads();

  // Gather knn_xyz [B,G,k,3]
  for (int p = tid; p < 16 * K_; p += 256) {
    int row = p / K_, j = p % K_;
    int nidx = s_topi[row][j];
    size_t dst = (((size_t)b * G_ + (g0 + row)) * K_ + j) * 3;
    size_t src = ((size_t)b * N_ + nidx) * 3;
    knn_xyz[dst + 0] = xyz[src + 0];
    knn_xyz[dst + 1] = xyz[src + 1];
    knn_xyz[dst + 2] = xyz[src + 2];
  }
  // Gather knn_x [B,G,k,C]: 64 threads per (row, neighbor) -> coalesced 256B rows
  int ch = tid & 63;
  for (int p = tid >> 6; p < 16 * K_; p += 4) {
    int row = p / K_, j = p % K_;
    int nidx = s_topi[row][j];
    knn_x[(((size_t)b * G_ + (g0 + row)) * K_ + j) * C_ + ch] =
        x[((size_t)b * N_ + nidx) * C_ + ch];
  }
}

// ------------------------------------------------------------------
// Host launcher. Output tuple concatenated flat:
//   lc_xyz [B,G,3] | lc_x [B,G,C] | knn_xyz [B,G,k,3] | knn_x [B,G,k,C]
// Workspace: fps indices (B*G ints = 16 KB).
// ------------------------------------------------------------------
extern "C" void kernel_launch(void* const* d_in, const int* in_sizes, int n_in,
                              void* d_out, int out_size, void* d_ws, size_t ws_size,
                              hipStream_t stream) {
  const float* xyz = (const float*)d_in[0];
  const float* x   = (const float*)d_in[1];
  (void)in_sizes; (void)n_in; (void)out_size; (void)ws_size;

  float* out     = (float*)d_out;
  float* lc_xyz  = out;                                   // B*G*3
  float* lc_x    = lc_xyz + (size_t)B_ * G_ * 3;          // B*G*C
  float* knn_xyz = lc_x   + (size_t)B_ * G_ * C_;         // B*G*K*3
  float* knn_x   = knn_xyz + (size_t)B_ * G_ * K_ * 3;    // B*G*K*C

  int* fps_idx = (int*)d_ws;

  fps_kernel<<<B_, 1024, 0, stream>>>(xyz, fps_idx);
  gather_kernel<<<B_ * G_, C_, 0, stream>>>(xyz, x, fps_idx, lc_xyz, lc_x);
  knn_kernel<<<dim3(G_ / 16, B_), 256, 0, stream>>>(xyz, x, lc_xyz, knn_xyz, knn_x);
}